// generatorGNN_57071525429606
// MI455X (gfx1250) — compile-verified
//
#include <hip/hip_runtime.h>
#include <hip/hip_bf16.h>

typedef __attribute__((ext_vector_type(2))) float v2f;
typedef __attribute__((ext_vector_type(8))) float v8f;

#define F_IN 128
#define H1   16
#define H2   32

// ---------------------------------------------------------------------------
// Degree / normalization
// ---------------------------------------------------------------------------
__global__ void k_init_deg(float* __restrict__ deg, int n) {
  int i = blockIdx.x * blockDim.x + threadIdx.x;
  if (i < n) deg[i] = 1.0f;  // self-loop weight
}

__global__ void k_deg_edges(const int* __restrict__ col, const float* __restrict__ w,
                            float* __restrict__ deg, int e) {
  int t = blockIdx.x * blockDim.x + threadIdx.x;
  if (t < e) atomicAdd(&deg[col[t]], w[t]);
}

__global__ void k_deg_to_dis(float* __restrict__ deg, int n) {
  int i = blockIdx.x * blockDim.x + threadIdx.x;
  if (i < n) {
    float d = deg[i];
    deg[i] = d > 0.0f ? rsqrtf(fmaxf(d, 1e-12f)) : 0.0f;
  }
}

__global__ void k_norm(const int* __restrict__ row, const int* __restrict__ col,
                       const float* __restrict__ w, const float* __restrict__ dis,
                       float* __restrict__ norm, int e) {
  int t = blockIdx.x * blockDim.x + threadIdx.x;
  if (t < e) norm[t] = dis[row[t]] * w[t] * dis[col[t]];
}

// ---------------------------------------------------------------------------
// GEMM1: h1[N,16] = x[N,128] @ W1[128,16]   (V_WMMA_F32_16X16X4_F32)
// Wave32 owns a 16-row x 16-col tile; K walked in steps of 4.
// A frag (16x4 f32): lane M = lane&15, K = 2*(lane>>4)+c  -> contiguous float2
// B frag (4x16 f32): lane N = lane&15, K = 2*(lane>>4)+c
// D (16x16 f32):     VGPR j -> M = j + 8*(lane>>4), N = lane&15
// ---------------------------------------------------------------------------
__global__ void k_gemm1_wmma(const float* __restrict__ x, const float* __restrict__ W1,
                             float* __restrict__ h1, int n) {
  __shared__ float w[F_IN * H1];  // 8 KB
  for (int i = threadIdx.x; i < F_IN * H1; i += 256) w[i] = W1[i];
  __syncthreads();

  const int wave = threadIdx.x >> 5;
  const int lane = threadIdx.x & 31;
  const int base = blockIdx.x * 128 + wave * 16;
  if (base >= n) return;  // wave-uniform: EXEC stays all-ones for WMMA

  const int m  = lane & 15;   // row of A, col of B/D
  const int kh = lane >> 4;   // K half-select
  const float* xrow = x + (size_t)(base + m) * F_IN + 2 * kh;

  v8f acc = {};
  for (int kk = 0; kk < F_IN; kk += 4) {
    v2f a = *(const v2f*)(xrow + kk);
    v2f b;
    b.x = w[(kk + 2 * kh) * H1 + m];
    b.y = w[(kk + 2 * kh + 1) * H1 + m];
    acc = __builtin_amdgcn_wmma_f32_16x16x4_f32(false, a, false, b, (short)0, acc,
                                                false, false);
  }
  float* o = h1 + (size_t)base * H1;
#pragma unroll
  for (int j = 0; j < 8; ++j) o[(j + 8 * kh) * H1 + m] = acc[j];
}

// ---------------------------------------------------------------------------
// GEMM2: h2[N,32] = h1a[N,16] @ W2[16,32]  (two 16-wide N tiles, shared A frag)
// ---------------------------------------------------------------------------
__global__ void k_gemm2_wmma(const float* __restrict__ h1a, const float* __restrict__ W2,
                             float* __restrict__ h2, int n) {
  __shared__ float w[H1 * H2];  // 2 KB
  for (int i = threadIdx.x; i < H1 * H2; i += 256) w[i] = W2[i];
  __syncthreads();

  const int wave = threadIdx.x >> 5;
  const int lane = threadIdx.x & 31;
  const int base = blockIdx.x * 128 + wave * 16;
  if (base >= n) return;

  const int m  = lane & 15;
  const int kh = lane >> 4;
  const float* arow = h1a + (size_t)(base + m) * H1 + 2 * kh;

  v8f acc0 = {}, acc1 = {};
#pragma unroll
  for (int kk = 0; kk < H1; kk += 4) {
    v2f a = *(const v2f*)(arow + kk);
    v2f b0, b1;
    b0.x = w[(kk + 2 * kh) * H2 + m];
    b0.y = w[(kk + 2 * kh + 1) * H2 + m];
    b1.x = w[(kk + 2 * kh) * H2 + 16 + m];
    b1.y = w[(kk + 2 * kh + 1) * H2 + 16 + m];
    acc0 = __builtin_amdgcn_wmma_f32_16x16x4_f32(false, a, false, b0, (short)0, acc0,
                                                 false, false);
    acc1 = __builtin_amdgcn_wmma_f32_16x16x4_f32(false, a, false, b1, (short)0, acc1,
                                                 false, false);
  }
  float* o = h2 + (size_t)base * H2;
#pragma unroll
  for (int j = 0; j < 8; ++j) {
    o[(j + 8 * kh) * H2 + m]      = acc0[j];
    o[(j + 8 * kh) * H2 + 16 + m] = acc1[j];
  }
}

// ---------------------------------------------------------------------------
// Aggregation: agg = dis^2 * h (self-loop init), then edge scatter-atomics.
// ---------------------------------------------------------------------------
template <int F>
__global__ void k_init_agg(const float* __restrict__ dis, const float* __restrict__ h,
                           float* __restrict__ agg, int n) {
  int t = blockIdx.x * blockDim.x + threadIdx.x;
  if (t < n * F) {
    int i = t / F;
    float d = dis[i];
    agg[t] = d * d * h[t];
  }
}

template <int F>
__global__ void k_scatter(const int* __restrict__ row, const int* __restrict__ col,
                          const float* __restrict__ norm, const float* __restrict__ h,
                          float* __restrict__ agg, int e) {
  long long t = (long long)blockIdx.x * blockDim.x + threadIdx.x;
  if (t >= (long long)e * F) return;
  int ed = (int)(t / F);  // F is a power of two -> shift
  int f  = (int)(t % F);
  int r = row[ed], c = col[ed];
  atomicAdd(&agg[(size_t)c * F + f], h[(size_t)r * F + f] * norm[ed]);
}

template <int F, bool RELU>
__global__ void k_bias_act(float* __restrict__ agg, const float* __restrict__ b, int n) {
  int t = blockIdx.x * blockDim.x + threadIdx.x;
  if (t < n * F) {
    float v = agg[t] + b[t % F];
    agg[t] = RELU ? fmaxf(v, 0.0f) : v;
  }
}

// ---------------------------------------------------------------------------
// Edge scorer factored through per-node scalars:
// out[e] = (h2[row]·Wc[0:32]) + (h2[col]·Wc[32:64]) + bc
// ---------------------------------------------------------------------------
__global__ void k_node_scores(const float* __restrict__ h2, const float* __restrict__ Wc,
                              float* __restrict__ s1, float* __restrict__ s2, int n) {
  int i = blockIdx.x * blockDim.x + threadIdx.x;
  if (i >= n) return;
  const float* r = h2 + (size_t)i * H2;
  float a = 0.0f, b = 0.0f;
#pragma unroll
  for (int k = 0; k < H2; ++k) {
    a += r[k] * Wc[k];
    b += r[k] * Wc[H2 + k];
  }
  s1[i] = a;
  s2[i] = b;
}

__global__ void k_edge_out(const int* __restrict__ row, const int* __restrict__ col,
                           const float* __restrict__ s1, const float* __restrict__ s2,
                           const float* __restrict__ bc, float* __restrict__ out, int e) {
  int t = blockIdx.x * blockDim.x + threadIdx.x;
  if (t < e) out[t] = s1[row[t]] + s2[col[t]] + bc[0];
}

// ---------------------------------------------------------------------------
extern "C" void kernel_launch(void* const* d_in, const int* in_sizes, int n_in,
                              void* d_out, int out_size, void* d_ws, size_t ws_size,
                              hipStream_t stream) {
  const float* x   = (const float*)d_in[0];
  const int*  eidx = (const int*)d_in[1];
  const float* ew  = (const float*)d_in[2];
  const float* W1  = (const float*)d_in[3];
  const float* b1  = (const float*)d_in[4];
  const float* W2  = (const float*)d_in[5];
  const float* b2  = (const float*)d_in[6];
  const float* Wc  = (const float*)d_in[7];
  const float* bc  = (const float*)d_in[8];
  float* out = (float*)d_out;

  const int N = in_sizes[0] / F_IN;
  const int E = in_sizes[1] / 2;
  const int* row = eidx;      // edge_index[0]
  const int* col = eidx + E;  // edge_index[1]

  // Workspace layout (all f32): ~52 MB total
  float* ws   = (float*)d_ws;
  float* dis  = ws;                           // [N]   deg -> dis (in place)
  float* norm = dis + N;                      // [E]
  float* h1   = norm + E;                     // [N*16]
  float* agg1 = h1 + (size_t)N * H1;          // [N*16] -> h1_act (in place)
  float* h2l  = agg1 + (size_t)N * H1;        // [N*32]
  float* agg2 = h2l + (size_t)N * H2;         // [N*32] -> h2 (in place)
  float* s1   = agg2 + (size_t)N * H2;        // [N]
  float* s2   = s1 + N;                       // [N]

  const int B = 256;
  auto blocks = [](long long total, int b) { return (int)((total + b - 1) / b); };

  // 1) degree + symmetric normalization
  k_init_deg<<<blocks(N, B), B, 0, stream>>>(dis, N);
  k_deg_edges<<<blocks(E, B), B, 0, stream>>>(col, ew, dis, E);
  k_deg_to_dis<<<blocks(N, B), B, 0, stream>>>(dis, N);
  k_norm<<<blocks(E, B), B, 0, stream>>>(row, col, ew, dis, norm, E);

  // 2) layer 1: WMMA transform, self-loop init, edge scatter, bias+ReLU
  k_gemm1_wmma<<<blocks(N, 128), B, 0, stream>>>(x, W1, h1, N);
  k_init_agg<H1><<<blocks((long long)N * H1, B), B, 0, stream>>>(dis, h1, agg1, N);
  k_scatter<H1><<<blocks((long long)E * H1, B), B, 0, stream>>>(row, col, norm, h1, agg1, E);
  k_bias_act<H1, true><<<blocks((long long)N * H1, B), B, 0, stream>>>(agg1, b1, N);

  // 3) layer 2
  k_gemm2_wmma<<<blocks(N, 128), B, 0, stream>>>(agg1, W2, h2l, N);
  k_init_agg<H2><<<blocks((long long)N * H2, B), B, 0, stream>>>(dis, h2l, agg2, N);
  k_scatter<H2><<<blocks((long long)E * H2, B), B, 0, stream>>>(row, col, norm, h2l, agg2, E);
  k_bias_act<H2, false><<<blocks((long long)N * H2, B), B, 0, stream>>>(agg2, b2, N);

  // 4) edge scores via per-node scalars (8 B/edge gather instead of 256 B)
  k_node_scores<<<blocks(N, B), B, 0, stream>>>(agg2, Wc, s1, s2, N);
  k_edge_out<<<blocks(E, B), B, 0, stream>>>(row, col, s1, s2, bc, out, E);
}